// GCN_Body_8237747274085
// MI455X (gfx1250) — compile-verified
//
#include <hip/hip_runtime.h>
#include <hip/hip_bf16.h>

#define NFEAT 256
#define NHID  64
#define BN_EPS 1e-5f
#define WT_STRIDE 260   // 256 + 4 pad -> conflict-free ds_load_b64 of B columns

typedef __attribute__((ext_vector_type(2))) float v2f;
typedef __attribute__((ext_vector_type(4))) float v4f;
typedef __attribute__((ext_vector_type(8))) float v8f;

// ---------------------------------------------------------------------------
// init: out = bias (folded), deg = 1 (self loop), stats = 0
// ---------------------------------------------------------------------------
__global__ void init_kernel(float* __restrict__ out, float* __restrict__ deg,
                            float* __restrict__ stats, const float* __restrict__ b,
                            int n_nodes) {
    int idx = blockIdx.x * blockDim.x + threadIdx.x;
    int total = n_nodes * NHID;
    if (idx < total)   out[idx] = b[idx & (NHID - 1)];
    if (idx < n_nodes) deg[idx] = 1.0f;          // self-loop contributes 1 to degree
    if (idx < 256)     stats[idx] = 0.0f;
}

// ---------------------------------------------------------------------------
// degree over targets (col)
// ---------------------------------------------------------------------------
__global__ void degree_kernel(const int* __restrict__ col, float* __restrict__ deg, int e_cnt) {
    int e = blockIdx.x * blockDim.x + threadIdx.x;
    if (e < e_cnt) atomicAdd(&deg[col[e]], 1.0f);
}

__global__ void dinv_kernel(float* __restrict__ deg, int n_nodes) {
    int i = blockIdx.x * blockDim.x + threadIdx.x;
    if (i < n_nodes) deg[i] = rsqrtf(deg[i]);    // deg >= 1 always
}

// ---------------------------------------------------------------------------
// GEMM h = x @ W via V_WMMA_F32_16X16X4_F32.
// Block = 256 threads = 8 waves; wave (rtile, ctile) computes a 16x16 tile.
// Block covers 32 rows x all 64 cols. W (256x64) is transposed into LDS
// (WldsT[n][k], stride 260) so each lane's B fragment is one 8-byte ds load.
// ---------------------------------------------------------------------------
__global__ void __launch_bounds__(256) gemm_wmma_kernel(
        const float* __restrict__ x, const float* __restrict__ W,
        float* __restrict__ h, int n_nodes) {
    __shared__ float WldsT[NHID * WT_STRIDE];

    const int tid = threadIdx.x;

    // cooperative load + transpose of W: 4096 float4 reads, coalesced
    const v4f* W4 = (const v4f*)W;
#pragma unroll
    for (int i = 0; i < 16; ++i) {
        int lin4 = i * 256 + tid;        // float4 index into W
        v4f v = W4[lin4];
        int base = lin4 * 4;
        int k = base >> 6;               // row of W   (0..255)
        int n = base & 63;               // col of W   (0..60, step 4)
        WldsT[(n + 0) * WT_STRIDE + k] = v.x;
        WldsT[(n + 1) * WT_STRIDE + k] = v.y;
        WldsT[(n + 2) * WT_STRIDE + k] = v.z;
        WldsT[(n + 3) * WT_STRIDE + k] = v.w;
    }
    __syncthreads();

    const int wave  = tid >> 5;
    const int lane  = tid & 31;
    const int rtile = wave >> 2;         // 0..1  (row tile within block)
    const int ctile = wave & 3;          // 0..3  (col tile)
    const int m     = lane & 15;         // row (A) / col (B) within tile
    const int half  = lane >> 4;         // K-half selector
    const int koff  = half * 2;

    long row = (long)blockIdx.x * 32 + rtile * 16 + m;
    long row_clamped = row < (long)n_nodes ? row : (long)n_nodes - 1; // keep EXEC all-1s
    const float* xr   = x + row_clamped * NFEAT;
    const float* bcol = &WldsT[(ctile * 16 + m) * WT_STRIDE];

    v8f c = {};
#pragma unroll 8
    for (int k = 0; k < NFEAT; k += 4) {
        // A 16x4 f32 fragment: v0/v1 = K={koff, koff+1} for this half-wave
        v2f a = *(const v2f*)(xr + k + koff);
        // B 4x16 f32 fragment: mirrored layout, 8B ds load, conflict-free
        v2f b = *(const v2f*)(bcol + k + koff);
        c = __builtin_amdgcn_wmma_f32_16x16x4_f32(
                /*neg_a=*/false, a, /*neg_b=*/false, b,
                /*c_mod=*/(short)0, c, /*reuse_a=*/false, /*reuse_b=*/false);
    }

    // D layout: VGPR r -> M = r + half*8, N = lane&15
    long rbase = (long)blockIdx.x * 32 + rtile * 16 + half * 8;
    int  ncol  = ctile * 16 + m;
#pragma unroll
    for (int r = 0; r < 8; ++r) {
        long grow = rbase + r;
        if (grow < (long)n_nodes)
            h[grow * NHID + ncol] = c[r];
    }
}

// ---------------------------------------------------------------------------
// scatter-aggregate: 64 threads per (edge | self-loop), coalesced h reads,
// global_atomic_add_f32 into out (which already holds the bias).
// ---------------------------------------------------------------------------
__global__ void edge_agg_kernel(const int* __restrict__ ei, const float* __restrict__ h,
                                const float* __restrict__ dinv, float* __restrict__ out,
                                int e_cnt, int n_nodes) {
    int gid = blockIdx.x * blockDim.x + threadIdx.x;
    int total = (e_cnt + n_nodes) * NHID;
    if (gid >= total) return;
    int e = gid >> 6;
    int f = gid & 63;
    int r, c;
    if (e < e_cnt) { r = ei[e]; c = ei[e_cnt + e]; }   // row = src, col = dst
    else           { r = c = e - e_cnt; }              // self loop
    float nrm = dinv[r] * dinv[c];
    atomicAdd(&out[(long)c * NHID + f], h[(long)r * NHID + f] * nrm);
}

// ---------------------------------------------------------------------------
// per-channel sum / sumsq reduction (block partials -> LDS -> 128 atomics/blk)
// ---------------------------------------------------------------------------
__global__ void __launch_bounds__(256) stats_kernel(const float* __restrict__ out,
                                                    float* __restrict__ stats, int n_nodes) {
    __shared__ float s0[256];
    __shared__ float s1[256];
    int tid = threadIdx.x;
    int f = tid & 63;                    // channel is fixed per thread
    int total = n_nodes * NHID;
    float sum = 0.0f, sq = 0.0f;
#pragma unroll
    for (int i = 0; i < 16; ++i) {
        int idx = (blockIdx.x * 16 + i) * 256 + tid;
        if (idx < total) {
            float v = out[idx];
            sum += v;
            sq  += v * v;
        }
    }
    s0[tid] = sum;
    s1[tid] = sq;
    __syncthreads();
    if (tid < 64) {
        sum = s0[tid] + s0[tid + 64] + s0[tid + 128] + s0[tid + 192];
        sq  = s1[tid] + s1[tid + 64] + s1[tid + 128] + s1[tid + 192];
        atomicAdd(&stats[f], sum);
        atomicAdd(&stats[64 + f], sq);
    }
}

__global__ void bn_finalize_kernel(float* __restrict__ stats,
                                   const float* __restrict__ bn_w,
                                   const float* __restrict__ bn_b, int n_nodes) {
    int f = threadIdx.x;
    if (f < NHID) {
        float inv_n = 1.0f / (float)n_nodes;
        float mean  = stats[f] * inv_n;
        float var   = stats[64 + f] * inv_n - mean * mean;   // biased variance
        float inv   = rsqrtf(var + BN_EPS);
        float scale = inv * bn_w[f];
        stats[128 + f] = scale;
        stats[192 + f] = bn_b[f] - mean * scale;
    }
}

__global__ void bn_apply_kernel(float* __restrict__ out,
                                const float* __restrict__ stats, int n_nodes) {
    int idx = blockIdx.x * blockDim.x + threadIdx.x;
    if (idx < n_nodes * NHID) {
        int f = idx & 63;
        out[idx] = out[idx] * stats[128 + f] + stats[192 + f];
    }
}

// ---------------------------------------------------------------------------
extern "C" void kernel_launch(void* const* d_in, const int* in_sizes, int n_in,
                              void* d_out, int out_size, void* d_ws, size_t ws_size,
                              hipStream_t stream) {
    const float* x    = (const float*)d_in[0];   // [N, 256]
    const int*   ei   = (const int*)d_in[1];     // [2, E]
    const float* W    = (const float*)d_in[2];   // [256, 64]
    const float* b    = (const float*)d_in[3];   // [64]
    const float* bn_w = (const float*)d_in[4];   // [64]
    const float* bn_b = (const float*)d_in[5];   // [64]
    float* out = (float*)d_out;                  // [N, 64]

    int n_nodes = in_sizes[0] / NFEAT;           // 100000
    int e_cnt   = in_sizes[1] / 2;               // 800000

    // workspace layout (floats): h [N*64] | deg/dinv [N] | stats [256]
    float* ws    = (float*)d_ws;
    float* h     = ws;
    float* deg   = ws + (size_t)n_nodes * NHID;
    float* stats = deg + n_nodes;

    int total = n_nodes * NHID;

    init_kernel<<<(total + 255) / 256, 256, 0, stream>>>(out, deg, stats, b, n_nodes);
    degree_kernel<<<(e_cnt + 255) / 256, 256, 0, stream>>>(ei + e_cnt, deg, e_cnt);
    dinv_kernel<<<(n_nodes + 255) / 256, 256, 0, stream>>>(deg, n_nodes);

    int gemm_blocks = (n_nodes + 31) / 32;
    gemm_wmma_kernel<<<gemm_blocks, 256, 0, stream>>>(x, W, h, n_nodes);

    int etotal = (e_cnt + n_nodes) * NHID;
    edge_agg_kernel<<<(etotal + 255) / 256, 256, 0, stream>>>(ei, h, deg, out, e_cnt, n_nodes);

    stats_kernel<<<(total + 4095) / 4096, 256, 0, stream>>>(out, stats, n_nodes);
    bn_finalize_kernel<<<1, 64, 0, stream>>>(stats, bn_w, bn_b, n_nodes);
    bn_apply_kernel<<<(total + 255) / 256, 256, 0, stream>>>(out, stats, n_nodes);
}